// TopKRoute_78305843740861
// MI455X (gfx1250) — compile-verified
//
#include <hip/hip_runtime.h>
#include <hip/hip_bf16.h>
#include <math.h>

typedef float v2f __attribute__((ext_vector_type(2)));
typedef float v8f __attribute__((ext_vector_type(8)));

#define KC   64            // K-chunk staged in LDS per iteration
#define PAD  4             // row padding (floats) -> conflict-free ds_load_2addr_b64
#define LDSW (KC + PAD)
#define NWG  512           // split-K workgroups
#define TPB  256           // 8 waves

// ---------------------------------------------------------------------------
// Split-K gate GEMM: acc[64,64] += x[64,K] @ W[64,K]^T  (fp32, WMMA 16x16x4)
// Double-buffered LDS stages filled by CDNA5 async global->LDS loads.
// ---------------------------------------------------------------------------
__global__ __launch_bounds__(TPB) void gate_gemm_kernel(
    const float* __restrict__ x, const float* __restrict__ W,
    float* __restrict__ acc, int K) {
  __shared__ float xs [2][64 * LDSW];
  __shared__ float wsh[2][64 * LDSW];

  const int tid  = threadIdx.x;
  const int lane = tid & 31;
  const int wave = tid >> 5;
  const int lr   = lane & 15;   // row within 16-wide tile
  const int half = lane >> 4;   // selects K pair {0,1} vs {2,3}

  const int mt  = wave >> 1;         // M tile 0..3 (batch rows)
  const int nt0 = (wave & 1) << 1;   // N tiles {0,1} or {2,3} (experts)

  v8f acc0 = {};
  v8f acc1 = {};

  // staging assignment: 4 threads per row, 16 contiguous floats (64 B) each
  const int srow = tid >> 2;
  const int sseg = (tid & 3) * 16;

  const unsigned long long xbase = (unsigned long long)x;
  const unsigned long long wbase = (unsigned long long)W;

  const int nChunks = K / KC;
  const int stride  = gridDim.x;

  // Issue async global->LDS loads of one 64xKC chunk of x and W into buffer b.
  // INST_OFFSET is added to BOTH the LDS address and the global address
  // (CDNA5 ISA), so one base VGPR each + offset:0/16/32/48
  // covers this thread's 64-byte slice.
  auto issue_chunk = [&](int b, int kbase) {
    const unsigned goff =
        ((unsigned)srow * (unsigned)K + (unsigned)(kbase + sseg)) * 4u;
    const unsigned ldsX = (unsigned)(uintptr_t)&xs [b][srow * LDSW + sseg];
    const unsigned ldsW = (unsigned)(uintptr_t)&wsh[b][srow * LDSW + sseg];
#pragma unroll
    for (int i = 0; i < 64; i += 16) {
      asm volatile("global_load_async_to_lds_b128 %0, %1, %2 offset:%3"
                   :: "v"(ldsX), "v"(goff), "s"(xbase), "i"(i) : "memory");
      asm volatile("global_load_async_to_lds_b128 %0, %1, %2 offset:%3"
                   :: "v"(ldsW), "v"(goff), "s"(wbase), "i"(i) : "memory");
    }
  };

  int buf = 0;
  if (blockIdx.x < nChunks) issue_chunk(0, blockIdx.x * KC);

  for (int chunk = blockIdx.x; chunk < nChunks; chunk += stride) {
    // Wait for this wave's async fills of buf, then WG-barrier so every
    // wave's portion of buf is visible before anyone reads it.
    asm volatile("s_wait_asynccnt 0x0" ::: "memory");
    __syncthreads();

    const int next = chunk + stride;
    if (next < nChunks) issue_chunk(buf ^ 1, next * KC);  // overlap with WMMAs

    const float* xrow  = &xs [buf][(mt * 16 + lr) * LDSW];
    const float* wrow0 = &wsh[buf][((nt0 + 0) * 16 + lr) * LDSW];
    const float* wrow1 = &wsh[buf][((nt0 + 1) * 16 + lr) * LDSW];

#pragma unroll
    for (int k0 = 0; k0 < KC; k0 += 4) {
      const int koff = k0 + 2 * half;
      // A 16x4 f32 frag: lane lr = M row, V0/V1 = K = koff/koff+1  (8B LDS load)
      v2f a  = *(const v2f*)(xrow + koff);
      // B 4x16 f32 frag: lane lr = N col, V0/V1 = K = koff/koff+1
      v2f b0 = *(const v2f*)(wrow0 + koff);
      v2f b1 = *(const v2f*)(wrow1 + koff);
      acc0 = __builtin_amdgcn_wmma_f32_16x16x4_f32(
          false, a, false, b0, (short)0, acc0, false, false);
      acc1 = __builtin_amdgcn_wmma_f32_16x16x4_f32(
          false, a, false, b1, (short)0, acc1, false, false);
    }
    __syncthreads();  // all reads of buf done before it is refilled
    buf ^= 1;
  }

  // C/D layout: VGPR v -> M = v + 8*half; lane lr -> N. Split-K reduce via f32 atomics.
#pragma unroll
  for (int v = 0; v < 8; ++v) {
    const int m  = mt * 16 + v + 8 * half;
    const int n0 = (nt0 + 0) * 16 + lr;
    const int n1 = (nt0 + 1) * 16 + lr;
    unsafeAtomicAdd(&acc[m * 64 + n0], acc0[v]);
    unsafeAtomicAdd(&acc[m * 64 + n1], acc1[v]);
  }
}

// ---------------------------------------------------------------------------
// Zero the split-K accumulator (workspace is poisoned by the harness)
// ---------------------------------------------------------------------------
__global__ void zero_acc_kernel(float* __restrict__ acc, int n) {
  int i = blockIdx.x * blockDim.x + threadIdx.x;
  if (i < n) acc[i] = 0.0f;
}

// ---------------------------------------------------------------------------
// Finalize: bias add, top-k select, softmax over scatter-into-zeros mask.
// One thread per batch row (64 rows). k read from device scalar.
// ---------------------------------------------------------------------------
__global__ __launch_bounds__(64) void finalize_kernel(
    const float* __restrict__ acc, const float* __restrict__ bias,
    const int* __restrict__ kptr, float* __restrict__ out) {
  const int row = threadIdx.x;  // 0..63
  float y[64];
#pragma unroll
  for (int j = 0; j < 64; ++j) y[j] = acc[row * 64 + j] + bias[j];

  int k = *kptr;
  if (k < 1) k = 1;
  if (k > 16) k = 16;

  unsigned long long sel = 0ull;
  float selval[16];
  for (int t = 0; t < k; ++t) {
    float best = -INFINITY;
    int bi = 0;
    for (int j = 0; j < 64; ++j) {
      if (!((sel >> j) & 1ull) && y[j] > best) { best = y[j]; bi = j; }
    }
    sel |= 1ull << bi;
    selval[t] = best;
  }

  // mask = {selval[..] at selected cols, 0 elsewhere}; softmax over all 64
  const float m = fmaxf(selval[0], 0.0f);
  float sum = (float)(64 - k) * expf(0.0f - m);
  for (int t = 0; t < k; ++t) sum += expf(selval[t] - m);
  const float inv = 1.0f / sum;
#pragma unroll
  for (int j = 0; j < 64; ++j) {
    const float mj = ((sel >> j) & 1ull) ? y[j] : 0.0f;
    out[row * 64 + j] = expf(mj - m) * inv;
  }
}

// ---------------------------------------------------------------------------
extern "C" void kernel_launch(void* const* d_in, const int* in_sizes, int n_in,
                              void* d_out, int out_size, void* d_ws, size_t ws_size,
                              hipStream_t stream) {
  const float* x  = (const float*)d_in[0];  // [64, 1024, 768] fp32
  const float* W  = (const float*)d_in[1];  // [64, 786432] fp32
  const float* b  = (const float*)d_in[2];  // [64] fp32
  const int*   kp = (const int*)d_in[3];    // scalar k

  const int n_exp = in_sizes[2];            // 64
  const int K     = in_sizes[1] / n_exp;    // 786432

  float* acc = (float*)d_ws;                // [64*64] split-K accumulator
  float* out = (float*)d_out;               // [64*64]

  zero_acc_kernel<<<(64 * 64 + 255) / 256, 256, 0, stream>>>(acc, 64 * 64);
  gate_gemm_kernel<<<NWG, TPB, 0, stream>>>(x, W, acc, K);
  finalize_kernel<<<1, 64, 0, stream>>>(acc, b, kp, out);
}